// PILayer_15745350107640
// MI455X (gfx1250) — compile-verified
//
#include <hip/hip_runtime.h>
#include <hip/hip_bf16.h>

typedef __attribute__((ext_vector_type(2))) float v2f;
typedef __attribute__((ext_vector_type(4))) float v4f;
typedef __attribute__((ext_vector_type(8))) float v8f;

#define N_CH      32
#define D_IN      64
#define H2        32
#define NB        8
#define K2TOT     256          // H2 * NB
#define W12T_STRIDE 66         // 64 + 2 pad (bank spread, 8B-aligned rows)
#define W3RT_STRIDE 258        // 256 + 2 pad
#define OFF_B12   2112         // 32 * 66
#define OFF_W3RT  2144         // + 32
#define WGT_FLOATS 10400       // 2144 + 32*258  (multiple of 4 -> b128 staging)
#define X2_STRIDE 36           // 32 + 4 pad: 16B-aligned rows, 64-bank coverage
#define X2_PER_WAVE 576        // 16 * 36
#define BAS_STRIDE 10          // 8 + 2 pad: 8B-aligned, bank-spread
#define BAS_PER_WAVE 160       // 16 * 10
#define OFF_X2    WGT_FLOATS
#define OFF_BAS   (WGT_FLOATS + 8 * X2_PER_WAVE)      // 15008
#define SMEM_FLOATS (OFF_BAS + 8 * BAS_PER_WAVE)      // 16288 -> 65152 B < 64KB

static __device__ __forceinline__ v8f wmma4(v2f a, v2f b, v8f c) {
  // V_WMMA_F32_16X16X4_F32 : D = A(16x4) x B(4x16) + C
  return __builtin_amdgcn_wmma_f32_16x16x4_f32(
      /*neg_a=*/false, a, /*neg_b=*/false, b,
      /*c_mod=*/(short)0, c, /*reuse_a=*/false, /*reuse_b=*/false);
}

// ---------------------------------------------------------------------------
// Prep: fuse W12 = W1@W2 (stored transposed+padded), b12 = b1@W2 + b2,
// and W3r^T[n][k*8+c] = W3[k][n*8+c] (transposed+padded). All into d_ws.
// ---------------------------------------------------------------------------
__global__ __launch_bounds__(256) void pil_prep(
    const float* __restrict__ W1, const float* __restrict__ b1,
    const float* __restrict__ W2, const float* __restrict__ b2,
    const float* __restrict__ W3, float* __restrict__ ws) {
  int t = threadIdx.x;
  // W12^T[n][k] = sum_j W1[k][j] * W2[j][n]
  for (int idx = t; idx < D_IN * H2; idx += 256) {
    int n = idx & 31, k = idx >> 5;
    float acc = 0.f;
    #pragma unroll 8
    for (int j = 0; j < 32; ++j) acc += W1[k * 32 + j] * W2[j * 32 + n];
    ws[n * W12T_STRIDE + k] = acc;
  }
  if (t < 32) {
    float acc = b2[t];
    #pragma unroll 8
    for (int j = 0; j < 32; ++j) acc += b1[j] * W2[j * 32 + t];
    ws[OFF_B12 + t] = acc;
  }
  // W3r^T[n][k2] = W3[k2>>3][n*8 + (k2&7)]
  for (int idx = t; idx < H2 * K2TOT; idx += 256) {
    int n = idx >> 8, k2 = idx & 255;
    ws[OFF_W3RT + n * W3RT_STRIDE + k2] = W3[(k2 >> 3) * 256 + n * 8 + (k2 & 7)];
  }
}

// ---------------------------------------------------------------------------
// Main: per wave, one 16-edge tile.
//   GEMM1: X[16,64](gathered) @ W12[64,32] + b12  -> X2[16,32]   (32 wmma)
//   GEMM2: Z[16,256] @ W3r[256,32] -> OUT[16,32], Z built on the fly (128 wmma)
// ---------------------------------------------------------------------------
__global__ __launch_bounds__(256) void pil_main(
    const float* __restrict__ prop, const int* __restrict__ idx_i,
    const int* __restrict__ idx_j, const float* __restrict__ basis,
    const float* __restrict__ fused, float* __restrict__ out, int E) {
  __shared__ float smem[SMEM_FLOATS];
  const int t = threadIdx.x;

  // stage fused weights into LDS, 16B-vectorized (shared by all 8 waves)
  for (int i = t * 4; i < WGT_FLOATS; i += 1024) {
    *(v4f*)(smem + i) = *(const v4f*)(fused + i);
  }

  const int lane = t & 31;
  const int wid  = t >> 5;
  const int half = lane >> 4;       // which half-wave (selects K-subcolumns)
  const int lm   = lane & 15;       // row (M) / col (N) within 16
  const int kk0  = half << 1;       // first of the two K elements this lane holds

  const long tile_base = ((long)blockIdx.x * 8 + wid) * 16;
  long e = tile_base + lm;
  long ecl = e < (long)(E - 1) ? e : (long)(E - 1);  // clamp loads for tail

  const int ni = idx_i[ecl];
  const int nj = idx_j[ecl];
  const float* pi = prop + (long)ni * N_CH;
  const float* pj = prop + (long)nj * N_CH;

  // Gather A-fragments for GEMM1: A(m=lm, k=4*kt+kk0 .. +1), directly in WMMA layout.
  v2f a[16];
  #pragma unroll
  for (int kt = 0; kt < 8; ++kt) {
    a[kt]     = *(const v2f*)(pi + 4 * kt + kk0);
    a[kt + 8] = *(const v2f*)(pj + 4 * kt + kk0);
  }

  // Stage this wave's basis rows into LDS (stride-10 rows: 8B-aligned, bank-spread)
  float* s_bas = smem + OFF_BAS + wid * BAS_PER_WAVE;
  {
    v4f bv = *(const v4f*)(basis + ecl * NB + 4 * half);
    *(v2f*)(s_bas + lm * BAS_STRIDE + 4 * half)     = __builtin_shufflevector(bv, bv, 0, 1);
    *(v2f*)(s_bas + lm * BAS_STRIDE + 4 * half + 2) = __builtin_shufflevector(bv, bv, 2, 3);
  }
  __syncthreads();

  const float* s_w12t = smem;
  const float* s_b12  = smem + OFF_B12;
  const float* s_w3rt = smem + OFF_W3RT;
  float*       s_x2   = smem + OFF_X2 + wid * X2_PER_WAVE;

  // ---- GEMM1: X2 = X @ W12 + b12 -----------------------------------------
  v8f c0 = {}, c1 = {};
  const float* w12a = s_w12t + lm * W12T_STRIDE + kk0;         // N = lm
  const float* w12b = s_w12t + (16 + lm) * W12T_STRIDE + kk0;  // N = 16+lm
  #pragma unroll
  for (int kt = 0; kt < 16; ++kt) {
    v2f b0 = *(const v2f*)(w12a + 4 * kt);
    v2f b1 = *(const v2f*)(w12b + 4 * kt);
    c0 = wmma4(a[kt], b0, c0);
    c1 = wmma4(a[kt], b1, c1);
  }
  const float bl0 = s_b12[lm];
  const float bl1 = s_b12[16 + lm];
  #pragma unroll
  for (int r = 0; r < 8; ++r) { c0[r] += bl0; c1[r] += bl1; }

  // Transpose X2 through LDS: C-layout (vgpr r, lane) -> row-major [m][k]
  #pragma unroll
  for (int r = 0; r < 8; ++r) {
    const int row = r + 8 * half;   // M = r (lanes 0-15) or r+8 (lanes 16-31)
    s_x2[row * X2_STRIDE + lm]      = c0[r];   // N = lm
    s_x2[row * X2_STRIDE + 16 + lm] = c1[r];   // N = 16+lm
  }
  __syncthreads();

  // Pull this lane's X2 row (m = lm) into registers as 8 x b128.
  const float* myx2 = s_x2 + lm * X2_STRIDE;
  v4f x2r[8];
  #pragma unroll
  for (int q = 0; q < 8; ++q) x2r[q] = *(const v4f*)(myx2 + 4 * q);

  // ---- GEMM2: OUT = Z @ W3r,  Z[m, k*8+c] = X2[m,k] * basis[m,c] ---------
  // k = kt2>>1 is half-independent (compile-time per unrolled iteration);
  // only c = 4*(kt2&1) + 2*half depends on the half-wave -> folded into mybh.
  v8f d0 = {}, d1 = {};
  const float* mybh = s_bas + lm * BAS_STRIDE + 2 * half;
  const float* wb0  = s_w3rt + lm * W3RT_STRIDE + kk0;         // N = lm
  const float* wb1  = s_w3rt + (16 + lm) * W3RT_STRIDE + kk0;  // N = 16+lm
  #pragma unroll
  for (int kt2 = 0; kt2 < 64; ++kt2) {
    const int   cbase = (kt2 & 1) * 4;
    const float xv    = x2r[kt2 >> 3][(kt2 >> 1) & 3];
    v2f bas = *(const v2f*)(mybh + cbase);
    v2f xv2; xv2[0] = xv; xv2[1] = xv;
    v2f av = xv2 * bas;                       // -> v_pk_mul_f32
    v2f b0 = *(const v2f*)(wb0 + 4 * kt2);
    v2f b1 = *(const v2f*)(wb1 + 4 * kt2);
    d0 = wmma4(av, b0, d0);
    d1 = wmma4(av, b1, d1);
  }

  // Store OUT[e, 0..31] from C-layout; predicate only the stores for the tail.
  #pragma unroll
  for (int r = 0; r < 8; ++r) {
    const long row = tile_base + r + 8 * half;
    if (row < (long)E) {
      out[row * N_CH + lm]      = d0[r];
      out[row * N_CH + 16 + lm] = d1[r];
    }
  }
}

extern "C" void kernel_launch(void* const* d_in, const int* in_sizes, int n_in,
                              void* d_out, int out_size, void* d_ws, size_t ws_size,
                              hipStream_t stream) {
  const float* prop  = (const float*)d_in[0];
  const int*   idx_i = (const int*)d_in[1];
  const int*   idx_j = (const int*)d_in[2];
  const float* basis = (const float*)d_in[3];
  const float* W1    = (const float*)d_in[4];
  const float* b1    = (const float*)d_in[5];
  const float* W2    = (const float*)d_in[6];
  const float* b2    = (const float*)d_in[7];
  const float* W3    = (const float*)d_in[8];
  float* out = (float*)d_out;
  float* ws  = (float*)d_ws;

  const int E = in_sizes[1];  // n_edges

  pil_prep<<<1, 256, 0, stream>>>(W1, b1, W2, b2, W3, ws);
  const int nblocks = (E + 127) / 128;  // 8 waves/block * 16 edges/wave
  pil_main<<<nblocks, 256, 0, stream>>>(prop, idx_i, idx_j, basis, ws, out, E);
}